// IRevNetSqueeze_52441550684474
// MI455X (gfx1250) — compile-verified
//
#include <hip/hip_runtime.h>
#include <hip/hip_bf16.h>

// Pixel-unshuffle (IRevNet squeeze), B=16, C=16, H=512, W=512, fp32.
//   out[b, 4c+2i+j, ho, wo] = x[b, c, 2ho+i, 2wo+j]
//
// Pure permutation: 0 FLOPs, 512 MB of HBM traffic -> entirely bandwidth
// bound (~22 us at 23.3 TB/s). Each thread streams 8 contiguous input floats
// (2x b128 NT loads), splits even/odd lanes in registers, and writes one
// float4 to each of the two destination channels (2x b128 NT stores).
// Every byte moves exactly once at 128-bit width, fully coalesced both ways.

#define B_  16
#define C_  16
#define H_  512
#define W_  512
#define HO_ (H_ / 2)
#define WO_ (W_ / 2)

// native clang vector: accepted by __builtin_nontemporal_* (HIP float4 is not)
typedef float v4f __attribute__((ext_vector_type(4)));

__global__ __launch_bounds__(256) void
irevnet_squeeze_kernel(const float* __restrict__ x, float* __restrict__ out) {
    // 8 input floats per thread
    const unsigned tid  = blockIdx.x * blockDim.x + threadIdx.x;
    const unsigned base = tid << 3;              // flat input index, W innermost

    const unsigned w0  = base & (W_ - 1);        // W = 512
    const unsigned row = base >> 9;              // (b*C + c)*H + h
    const unsigned h   = row & (H_ - 1);
    const unsigned bc  = row >> 9;               // b*C + c
    const unsigned c   = bc & (C_ - 1);
    const unsigned b   = bc >> 4;

    const unsigned i  = h & 1u;
    const unsigned ho = h >> 1;
    const unsigned co = (c << 2) | (i << 1);     // output channel for j=0

    // 2x 16B contiguous non-temporal loads (32B per lane, coalesced)
    const v4f* in4 = reinterpret_cast<const v4f*>(x + base);
    const v4f a = __builtin_nontemporal_load(in4);
    const v4f d = __builtin_nontemporal_load(in4 + 1);

    v4f even;  // j = 0 -> channel co
    even.x = a.x; even.y = a.z; even.z = d.x; even.w = d.z;
    v4f odd;   // j = 1 -> channel co + 1
    odd.x  = a.y; odd.y  = a.w; odd.z  = d.y; odd.w  = d.w;

    // out flat index: ((b*4C + co)*HO + ho)*WO + w0/2
    const unsigned plane0 = (b << 6) | co;       // b*64 + co
    const unsigned obase  = ((plane0 * HO_ + ho) * WO_) + (w0 >> 1);

    v4f* o0 = reinterpret_cast<v4f*>(out + obase);              // channel co
    v4f* o1 = reinterpret_cast<v4f*>(out + obase + HO_ * WO_);  // channel co+1
    __builtin_nontemporal_store(even, o0);
    __builtin_nontemporal_store(odd,  o1);
}

extern "C" void kernel_launch(void* const* d_in, const int* in_sizes, int n_in,
                              void* d_out, int out_size, void* d_ws, size_t ws_size,
                              hipStream_t stream) {
    (void)in_sizes; (void)n_in; (void)d_ws; (void)ws_size; (void)out_size;
    const float* x  = (const float*)d_in[0];
    float* out      = (float*)d_out;

    const unsigned total_threads = (B_ * C_ * H_ * W_) / 8;  // 8,388,608
    const unsigned block = 256;                              // 8 wave32s
    const unsigned grid  = total_threads / block;            // 32,768
    irevnet_squeeze_kernel<<<grid, block, 0, stream>>>(x, out);
}